// LateralInhibition_4999341933025
// MI455X (gfx1250) — compile-verified
//
#include <hip/hip_runtime.h>

// ---------------------------------------------------------------------------
// Top-k lateral inhibition via MSB-first radix select.
// B=8 batches, N elements each.  Workspace layout (u32 words):
//   [0, 8192)            : 4 passes x (8 batches x 256 bins) histograms
//   [8192, 8208)         : per-batch state {prefix/threshold, kprime/E}
//   [8208, 8208+8192)    : per-block equal-counts (later: exclusive prefixes)
// ---------------------------------------------------------------------------
#define HIST_WORDS (4 * 2048)
#define STATE_OFF  HIST_WORDS
#define BLKEQ_OFF  (HIST_WORDS + 16)

// monotonic mapping: larger float  <=>  larger u32
__device__ __forceinline__ unsigned mono(float f) {
  unsigned u = __float_as_uint(f);
  return (u & 0x80000000u) ? ~u : (u | 0x80000000u);
}

// ---- CDNA5 async global->LDS DMA (ASYNCcnt path) --------------------------
__device__ __forceinline__ void async_copy_b128(void* lds_dst, const void* gsrc) {
  // generic LDS pointer: low 32 bits are the LDS byte offset (aperture rule)
  unsigned lds_off = (unsigned)(unsigned long long)lds_dst;
  asm volatile("global_load_async_to_lds_b128 %0, %1, off"
               :: "v"(lds_off), "v"((unsigned long long)gsrc)
               : "memory");
}
#if __has_builtin(__builtin_amdgcn_s_wait_asynccnt)
  #define WAIT_ASYNC(n) __builtin_amdgcn_s_wait_asynccnt(n)
#else
  #define WAIT_ASYNC(n) asm volatile("s_wait_asynccnt %0" :: "i"(n) : "memory")
#endif

// ---------------------------------------------------------------------------
// init: zero histograms, seed per-batch state {prefix=0, kprime=k}
// ---------------------------------------------------------------------------
__global__ void init_ws(unsigned* __restrict__ W, unsigned k) {
  int i = blockIdx.x * blockDim.x + threadIdx.x;
  if (i < HIST_WORDS) W[i] = 0u;
  if (i < 8) { W[STATE_OFF + 2 * i] = 0u; W[STATE_OFF + 2 * i + 1] = k; }
}

// ---------------------------------------------------------------------------
// histogram pass p (p = 0..3): 256-bin LDS histogram of byte (m >> (24-8p)),
// counting only elements whose higher bytes match the current prefix.
// Data is streamed through LDS with double-buffered async DMA.
// Block = 256 threads (8 wave32), covers 4096 contiguous elements.
// ---------------------------------------------------------------------------
__global__ void __launch_bounds__(256)
hist_pass(const float* __restrict__ x, unsigned* __restrict__ W,
          int p, int blocksPerBatch) {
  __shared__ unsigned hist[256];
  __shared__ float4 stage[2][256];

  const int t     = threadIdx.x;
  const int blk   = blockIdx.x;
  const int batch = blk / blocksPerBatch;

  hist[t] = 0u;
  const unsigned prefix  = W[STATE_OFF + 2 * batch];
  const int      shift   = 24 - 8 * p;
  const unsigned topmask = p ? (0xFFFFFFFFu << ((shift + 8) & 31)) : 0u;

  const float4* g = (const float4*)x + (long long)blk * 1024 + t;
  __syncthreads();

  // prime the pipeline: iteration 0 tile
  async_copy_b128(&stage[0][t], g);

  #pragma unroll
  for (int it = 0; it < 4; ++it) {
    const int cur = it & 1;
    if (it < 3) {
      async_copy_b128(&stage[cur ^ 1][t], g + (it + 1) * 256);
      WAIT_ASYNC(1);            // current tile done, next still in flight
    } else {
      WAIT_ASYNC(0);            // drain
    }
    float4 v = stage[cur][t];   // each lane consumes its own staged slot
    float c4[4] = {v.x, v.y, v.z, v.w};
    #pragma unroll
    for (int j = 0; j < 4; ++j) {
      unsigned m = mono(c4[j]);
      if (((m ^ prefix) & topmask) == 0u)
        atomicAdd(&hist[(m >> shift) & 0xFFu], 1u);
    }
  }

  __syncthreads();
  unsigned c = hist[t];
  if (c) atomicAdd(&W[(unsigned)p * 2048u + (unsigned)batch * 256u + (unsigned)t], c);
}

// ---------------------------------------------------------------------------
// per-batch serial bin selection (256 iterations, 1 thread per batch)
// ---------------------------------------------------------------------------
__global__ void select_bin(unsigned* __restrict__ W, int p) {
  const int b = threadIdx.x;                       // 0..7
  const unsigned* h  = W + (unsigned)p * 2048u + (unsigned)b * 256u;
  unsigned*       st = W + STATE_OFF + 2 * b;
  unsigned kprime = st[1];
  unsigned cum = 0; int c = 0;
  for (int bin = 255; bin >= 0; --bin) {
    unsigned cnt = h[bin];
    if (cum + cnt >= kprime) { c = bin; break; }
    cum += cnt;
  }
  st[0] |= (unsigned)c << (24 - 8 * p);            // extend prefix
  st[1]  = kprime - cum;                           // rank within chosen bin
}

// ---------------------------------------------------------------------------
// per-block count of elements equal to the exact threshold T
// thread t owns contiguous elements [t*16, t*16+16) of its block (index order)
// ---------------------------------------------------------------------------
__global__ void __launch_bounds__(256)
count_eq(const float* __restrict__ x, unsigned* __restrict__ W, int blocksPerBatch) {
  const int t = threadIdx.x, blk = blockIdx.x;
  const int batch = blk / blocksPerBatch;
  const unsigned T = W[STATE_OFF + 2 * batch];
  const float4* g = (const float4*)x + (long long)blk * 1024 + t * 4;
  unsigned c = 0;
  #pragma unroll
  for (int i = 0; i < 4; ++i) {
    float4 v = g[i];
    c += (mono(v.x) == T) + (mono(v.y) == T) + (mono(v.z) == T) + (mono(v.w) == T);
  }
  __shared__ unsigned s[256];
  s[t] = c; __syncthreads();
  for (int o = 128; o > 0; o >>= 1) { if (t < o) s[t] += s[t + o]; __syncthreads(); }
  if (t == 0) W[BLKEQ_OFF + blk] = s[0];
}

// ---------------------------------------------------------------------------
// per-batch exclusive scan over the 1024 block equal-counts (one WG per batch)
// ---------------------------------------------------------------------------
__global__ void __launch_bounds__(1024)
scan_blocks(unsigned* __restrict__ W, int blocksPerBatch) {
  __shared__ unsigned s[1024];
  const int b = blockIdx.x, t = threadIdx.x;
  unsigned* be = W + BLKEQ_OFF + b * blocksPerBatch;
  unsigned v = (t < blocksPerBatch) ? be[t] : 0u;
  s[t] = v; __syncthreads();
  for (int o = 1; o < (int)blockDim.x; o <<= 1) {
    unsigned add = (t >= (int)o) ? s[t - o] : 0u;
    __syncthreads();
    s[t] += add;
    __syncthreads();
  }
  if (t < blocksPerBatch) be[t] = s[t] - v;        // exclusive
}

// ---------------------------------------------------------------------------
// final mask: keep x if m > T; among m == T keep the first E by global index
// (block prefix + intra-block scan + per-thread sequential rank)
// ---------------------------------------------------------------------------
__global__ void __launch_bounds__(256)
mask_pass(const float* __restrict__ x, const unsigned* __restrict__ W,
          float* __restrict__ out, int blocksPerBatch) {
  const int t = threadIdx.x, blk = blockIdx.x;
  const int batch = blk / blocksPerBatch;
  const unsigned T = W[STATE_OFF + 2 * batch];
  const unsigned E = W[STATE_OFF + 2 * batch + 1];
  const unsigned blockBase = W[BLKEQ_OFF + blk];

  const long long base4 = (long long)blk * 1024 + t * 4;
  const float4* g = (const float4*)x + base4;
  float4 v[4]; unsigned m[16]; unsigned myeq = 0;
  #pragma unroll
  for (int i = 0; i < 4; ++i) {
    v[i] = g[i];
    float c4[4] = {v[i].x, v[i].y, v[i].z, v[i].w};
    #pragma unroll
    for (int j = 0; j < 4; ++j) { m[i * 4 + j] = mono(c4[j]); myeq += (m[i * 4 + j] == T); }
  }

  __shared__ unsigned s[256];
  s[t] = myeq; __syncthreads();
  for (int o = 1; o < 256; o <<= 1) {
    unsigned add = (t >= o) ? s[t - o] : 0u;
    __syncthreads();
    s[t] += add;
    __syncthreads();
  }
  unsigned rank = blockBase + (s[t] - myeq);       // global tie rank of my 1st eq

  float4* o4 = (float4*)out + base4;
  #pragma unroll
  for (int i = 0; i < 4; ++i) {
    float r[4]; float c4[4] = {v[i].x, v[i].y, v[i].z, v[i].w};
    #pragma unroll
    for (int j = 0; j < 4; ++j) {
      unsigned mm = m[i * 4 + j];
      float val = c4[j], keep = 0.0f;
      if (mm > T) keep = val;
      else if (mm == T) { keep = (rank < E) ? val : 0.0f; rank++; }
      r[j] = keep;
    }
    o4[i] = make_float4(r[0], r[1], r[2], r[3]);
  }
}

// ---------------------------------------------------------------------------
extern "C" void kernel_launch(void* const* d_in, const int* in_sizes, int n_in,
                              void* d_out, int out_size, void* d_ws, size_t ws_size,
                              hipStream_t stream) {
  const float* x = (const float*)d_in[0];
  float* out = (float*)d_out;
  unsigned* W = (unsigned*)d_ws;

  const long long total = in_sizes[0];             // 8*32*32*64*64 = 33,554,432
  const int B = 8;
  const long long N = total / B;                   // 4,194,304 per batch
  const int blocksPerBatch = (int)(N / 4096);      // 1024
  const int totalBlocks = B * blocksPerBatch;      // 8192
  unsigned k = (unsigned)(0.1 * (double)N);        // int(K_FRACTION * N)
  if (k < 1u) k = 1u;

  init_ws<<<32, 256, 0, stream>>>(W, k);
  for (int p = 0; p < 4; ++p) {
    hist_pass<<<totalBlocks, 256, 0, stream>>>(x, W, p, blocksPerBatch);
    select_bin<<<1, B, 0, stream>>>(W, p);
  }
  count_eq<<<totalBlocks, 256, 0, stream>>>(x, W, blocksPerBatch);
  scan_blocks<<<B, blocksPerBatch, 0, stream>>>(W, blocksPerBatch);
  mask_pass<<<totalBlocks, 256, 0, stream>>>(x, W, out, blocksPerBatch);
}